// VanillaClassifierStableV1_34471407518047
// MI455X (gfx1250) — compile-verified
//
#include <hip/hip_runtime.h>

// ---------------------------------------------------------------------------
// Cross-entropy (log_softmax + gather + mean) for B=4096, C=50257 fp32 logits.
// Memory-bound: 823 MB single pass @ 23.3 TB/s ~= 35us floor.
// Kernel 1: one block per row, streaming online softmax (1 read per element),
//           b128 non-temporal loads (no reuse -> don't pollute caches).
// Kernel 2: deterministic 4096 -> 1 reduction using V_WMMA_F32_16X16X4_F32.
// ---------------------------------------------------------------------------

typedef __attribute__((ext_vector_type(2))) float v2f;
typedef __attribute__((ext_vector_type(4))) float v4f;
typedef __attribute__((ext_vector_type(8))) float v8f;

#define L2E 1.44269504088896340736f   // log2(e)
#define LN2 0.69314718055994530942f   // ln(2)

__device__ __forceinline__ void online_accum(float x, float& m, float& s) {
    // Running max m, running sum s = sum 2^((x - m) * log2e).
    // Common case (x <= m): one v_exp_f32. Rare case rescales.
    if (x <= m) {
        s += exp2f((x - m) * L2E);
    } else {
        s = s * exp2f((m - x) * L2E) + 1.0f;
        m = x;
    }
}

__global__ __launch_bounds__(256) void ce_row_kernel(
    const float* __restrict__ pred,
    const long long* __restrict__ target,
    float* __restrict__ row_loss,
    int C)
{
    const int row = blockIdx.x;
    const int tid = threadIdx.x;
    const int C4  = C >> 2;                       // float4 count
    const float* rowp = pred + (long long)row * (long long)C;
    const v4f* rowp4 = (const v4f*)rowp;

    float m = -3.402823466e38f;                   // -FLT_MAX
    float s = 0.0f;

    // Streaming pass: coalesced b128 non-temporal loads.
    for (int i = tid; i < C4; i += 256) {
        v4f v = __builtin_nontemporal_load(&rowp4[i]);
        online_accum(v.x, m, s);
        online_accum(v.y, m, s);
        online_accum(v.z, m, s);
        online_accum(v.w, m, s);
    }
    // Tail elements (C % 4)
    for (int j = (C4 << 2) + tid; j < C; j += 256) {
        online_accum(rowp[j], m, s);
    }

    // Wave32 reduction of (m, s) pairs.
    for (int off = 16; off >= 1; off >>= 1) {
        float om = __shfl_xor(m, off, 32);
        float os = __shfl_xor(s, off, 32);
        float nm = fmaxf(m, om);
        s = s * exp2f((m - nm) * L2E) + os * exp2f((om - nm) * L2E);
        m = nm;
    }

    __shared__ float sm[8];
    __shared__ float ss[8];
    const int wave = tid >> 5;
    const int lane = tid & 31;
    if (lane == 0) { sm[wave] = m; ss[wave] = s; }
    __syncthreads();

    if (tid == 0) {
        float M = sm[0], S = ss[0];
        #pragma unroll
        for (int w = 1; w < 8; ++w) {
            float nm = fmaxf(M, sm[w]);
            S = S * exp2f((M - nm) * L2E) + ss[w] * exp2f((sm[w] - nm) * L2E);
            M = nm;
        }
        const long long t = target[row];
        const float xt = rowp[t];
        // loss = ln(sum exp(x - M)) + M - x_t ; sum kept base-2
        row_loss[row] = log2f(S) * LN2 + (M - xt);
    }
}

// Deterministic 4096 -> 1 sum using V_WMMA_F32_16X16X4_F32 as an f32 reducer.
// B = ones => D[m][n] accumulates row-sums of A; 8 waves x 8 chained WMMAs
// each fold 512 values in full fp32.
__global__ __launch_bounds__(256) void final_reduce_wmma(
    const float* __restrict__ row_loss,
    float* __restrict__ out,
    int n)
{
    const int tid  = threadIdx.x;
    const int wave = tid >> 5;
    const int lane = tid & 31;
    __shared__ float wsum[8];

    v8f c = {};                       // f32 accumulator (16x16)
    v2f ones; ones[0] = 1.0f; ones[1] = 1.0f;

    const int base = wave * 512;
    #pragma unroll
    for (int i = 0; i < 8; ++i) {
        const int i0 = base + i * 64 + lane * 2;
        v2f a;
        a[0] = (i0 + 0 < n) ? row_loss[i0 + 0] : 0.0f;
        a[1] = (i0 + 1 < n) ? row_loss[i0 + 1] : 0.0f;
        // D = A(16x4) * ones(4x16) + C  -> per-row sums accumulate in C
        c = __builtin_amdgcn_wmma_f32_16x16x4_f32(
            /*neg_a=*/false, a, /*neg_b=*/false, ones,
            /*c_mod=*/(short)0, c, /*reuse_a=*/false, /*reuse_b=*/false);
    }

    // Lane L (0..15) column holds M=0..7 in c[0..7]; lane L+16 holds M=8..15.
    float t = c[0] + c[1] + c[2] + c[3] + c[4] + c[5] + c[6] + c[7];
    t += __shfl_xor(t, 16, 32);       // full 512-value sum in every lane
    if (lane == 0) wsum[wave] = t;
    __syncthreads();

    if (tid == 0) {
        float total = 0.0f;
        #pragma unroll
        for (int w = 0; w < 8; ++w) total += wsum[w];
        out[0] = -total / (float)n;
    }
}

extern "C" void kernel_launch(void* const* d_in, const int* in_sizes, int n_in,
                              void* d_out, int out_size, void* d_ws, size_t ws_size,
                              hipStream_t stream) {
    const float*     pred   = (const float*)d_in[0];
    const long long* target = (const long long*)d_in[1];
    float* out = (float*)d_out;
    float* ws  = (float*)d_ws;       // per-row losses: B floats

    const int B = in_sizes[1];                       // 4096
    const int C = (int)(in_sizes[0] / (long long)B); // 50257

    ce_row_kernel<<<B, 256, 0, stream>>>(pred, target, ws, C);
    final_reduce_wmma<<<1, 256, 0, stream>>>(ws, out, B);
}